// LayerG_34926674051409
// MI455X (gfx1250) — compile-verified
//
#include <hip/hip_runtime.h>
#include <math.h>

#define N_NODES 50000
#define N_EDGES 600000
#define D 128
#define REL_DIM 100
#define N_REL 40

typedef float v2f __attribute__((ext_vector_type(2)));
typedef float v8f __attribute__((ext_vector_type(8)));

// Order-preserving float <-> uint mapping for atomicMax on floats.
__device__ __forceinline__ unsigned ord_key(float f) {
  unsigned u = __float_as_uint(f);
  return (u & 0x80000000u) ? ~u : (u | 0x80000000u);
}
__device__ __forceinline__ float ord_val(unsigned k) {
  unsigned u = (k & 0x80000000u) ? (k ^ 0x80000000u) : ~k;
  return __uint_as_float(u);
}

// ---------------- init: zero agg / node_sum, node_max = key(-inf) -------------
__global__ void init_kernel(float* __restrict__ agg, float* __restrict__ node_sum,
                            unsigned* __restrict__ node_max, int nd, int n) {
  int i = blockIdx.x * blockDim.x + threadIdx.x;
  int stride = gridDim.x * blockDim.x;
  for (int j = i; j < nd; j += stride) agg[j] = 0.f;
  for (int j = i; j < n; j += stride) { node_sum[j] = 0.f; node_max[j] = 0x007FFFFFu; }
}

// -------- per-relation logit table: rel_logit[r] = (rel_emb[r] @ W_r) . a[2D:]
__global__ void rel_kernel(const float* __restrict__ W_r, const float* __restrict__ a,
                           const float* __restrict__ rel_emb, float* __restrict__ rel_logit) {
  __shared__ float v[REL_DIM];
  int t = threadIdx.x;  // 128 threads
  if (t < REL_DIM) {
    float s = 0.f;
    for (int k = 0; k < D; ++k) s += W_r[t * D + k] * a[2 * D + k];
    v[t] = s;
  }
  __syncthreads();
  if (t < N_REL) {
    float s = 0.f;
    for (int j = 0; j < REL_DIM; ++j) s += rel_emb[t * REL_DIM + j] * v[j];
    rel_logit[t] = s;
  }
}

// -------- fused dual GEMM via V_WMMA_F32_16X16X4_F32:
//          h = X @ W    and    base = X @ W_res + b_res
// One wave computes a 16x128 strip of both outputs; 4 waves / block.
__global__ void __launch_bounds__(128) gemm_kernel(
    const float* __restrict__ X, const float* __restrict__ W,
    const float* __restrict__ Wres, const float* __restrict__ bres,
    float* __restrict__ h, float* __restrict__ base) {
  const int wave = threadIdx.x >> 5;
  const int lane = threadIdx.x & 31;
  const int row0 = (blockIdx.x * 4 + wave) * 16;
  if (row0 >= N_NODES) return;          // wave-uniform: WMMA keeps EXEC all-ones
  const int m  = lane & 15;             // A: row M ; B: col N
  const int kh = (lane >> 4) * 2;       // K sub-offset within group of 4

  v8f accW[8] = {};
  v8f accR[8] = {};

  const float* ap = X + (size_t)(row0 + m) * D + kh;
  for (int k0 = 0; k0 < D; k0 += 4) {
    v2f afrag;
    afrag.x = ap[k0];
    afrag.y = ap[k0 + 1];
    const float* wrow = W    + (size_t)(k0 + kh) * D + m;
    const float* rrow = Wres + (size_t)(k0 + kh) * D + m;
#pragma unroll
    for (int t = 0; t < 8; ++t) {
      v2f bw; bw.x = wrow[t * 16]; bw.y = wrow[t * 16 + D];
      accW[t] = __builtin_amdgcn_wmma_f32_16x16x4_f32(
          false, afrag, false, bw, (short)0, accW[t], false, false);
      v2f br; br.x = rrow[t * 16]; br.y = rrow[t * 16 + D];
      accR[t] = __builtin_amdgcn_wmma_f32_16x16x4_f32(
          false, afrag, false, br, (short)0, accR[t], false, false);
    }
  }

  // C/D layout: VGPR v, lanes 0-15 -> M=v, N=lane ; lanes 16-31 -> M=v+8, N=lane-16
  const int rbase = row0 + ((lane >> 4) << 3);
#pragma unroll
  for (int t = 0; t < 8; ++t) {
    float bb = bres[t * 16 + m];
#pragma unroll
    for (int v = 0; v < 8; ++v) {
      size_t idx = (size_t)(rbase + v) * D + t * 16 + m;
      h[idx]    = accW[t][v];
      base[idx] = accR[t][v] + bb;
    }
  }
}

// -------- per-node attention scalars: s_dst = h.a[:D], s_src = h.a[D:2D]
__global__ void __launch_bounds__(256) s_kernel(const float* __restrict__ h,
                                                const float* __restrict__ a,
                                                float* __restrict__ s_dst,
                                                float* __restrict__ s_src) {
  int node = blockIdx.x * (blockDim.x >> 5) + (threadIdx.x >> 5);
  int lane = threadIdx.x & 31;
  if (node >= N_NODES) return;
  const float4* hp = (const float4*)(h + (size_t)node * D);
  float4 x = hp[lane];
  const float4* ad = (const float4*)a;
  float4 w0 = ad[lane];        // a[:128]
  float4 w1 = ad[32 + lane];   // a[128:256]
  float d0 = x.x * w0.x + x.y * w0.y + x.z * w0.z + x.w * w0.w;
  float d1 = x.x * w1.x + x.y * w1.y + x.z * w1.z + x.w * w1.w;
#pragma unroll
  for (int off = 16; off > 0; off >>= 1) {
    d0 += __shfl_xor(d0, off, 32);
    d1 += __shfl_xor(d1, off, 32);
  }
  if (lane == 0) { s_dst[node] = d0; s_src[node] = d1; }
}

// -------- pass 1 over edges: leaky-relu logit + segment max (atomic on keys)
__global__ void logit_kernel(const int* __restrict__ src, const int* __restrict__ dst,
                             const int* __restrict__ etype,
                             const float* __restrict__ s_dst, const float* __restrict__ s_src,
                             const float* __restrict__ rel_logit,
                             float* __restrict__ elog, unsigned* __restrict__ node_max) {
  int i = blockIdx.x * blockDim.x + threadIdx.x;
  if (i >= N_EDGES) return;
  int d = dst[i];
  float l = s_dst[d] + s_src[src[i]] + rel_logit[etype[i]];
  l = (l > 0.f) ? l : 0.2f * l;
  elog[i] = l;
  atomicMax(node_max + d, ord_key(l));
}

// -------- pass 2: e = exp(logit - max[dst]); segment sum
__global__ void expsum_kernel(const int* __restrict__ dst, float* __restrict__ elog,
                              const unsigned* __restrict__ node_max,
                              float* __restrict__ node_sum) {
  int i = blockIdx.x * blockDim.x + threadIdx.x;
  if (i >= N_EDGES) return;
  int d = dst[i];
  float e = __expf(elog[i] - ord_val(node_max[d]));
  elog[i] = e;
  atomicAdd(node_sum + d, e);
}

// -------- pass 3: agg[dst] += alpha * h[src]; one wave per edge, coalesced atomics
__global__ void __launch_bounds__(256) scatter_kernel(
    const int* __restrict__ src, const int* __restrict__ dst,
    const float* __restrict__ elog, const float* __restrict__ node_sum,
    const float* __restrict__ h, float* __restrict__ agg) {
  int e = blockIdx.x * (blockDim.x >> 5) + (threadIdx.x >> 5);
  int lane = threadIdx.x & 31;
  if (e >= N_EDGES) return;
  int s = src[e], d = dst[e];
  float alpha = elog[e] / (node_sum[d] + 1e-16f);
  const float* hs = h + (size_t)s * D;
  float* ag = agg + (size_t)d * D;
#pragma unroll
  for (int j = 0; j < 4; ++j) {
    int c = lane + 32 * j;                 // fully coalesced 128B atomic bursts
    atomicAdd(ag + c, hs[c] * alpha);
  }
}

// -------- finalize: out = selu(agg + base)
__global__ void final_kernel(const float* __restrict__ agg, const float* __restrict__ base,
                             float* __restrict__ out) {
  int i = blockIdx.x * blockDim.x + threadIdx.x;
  if (i >= N_NODES * D) return;
  const float SCALE = 1.0507009873554805f, ALPHA = 1.6732632423543772f;
  float x = agg[i] + base[i];
  out[i] = (x > 0.f) ? SCALE * x : SCALE * ALPHA * (__expf(x) - 1.f);
}

extern "C" void kernel_launch(void* const* d_in, const int* in_sizes, int n_in,
                              void* d_out, int out_size, void* d_ws, size_t ws_size,
                              hipStream_t stream) {
  const float* X        = (const float*)d_in[0];
  const int*   eidx     = (const int*)d_in[1];
  const int*   etype    = (const int*)d_in[2];
  const float* W        = (const float*)d_in[3];
  const float* W_r      = (const float*)d_in[4];
  const float* a        = (const float*)d_in[5];
  const float* Wres     = (const float*)d_in[6];
  const float* bres     = (const float*)d_in[7];
  const float* rel_emb  = (const float*)d_in[8];
  const int* src = eidx;             // edge_index[0]
  const int* dst = eidx + N_EDGES;   // edge_index[1]

  float* ws = (float*)d_ws;
  float*    h         = ws;              ws += (size_t)N_NODES * D;
  float*    base      = ws;              ws += (size_t)N_NODES * D;
  float*    agg       = ws;              ws += (size_t)N_NODES * D;
  float*    s_dst     = ws;              ws += N_NODES;
  float*    s_src     = ws;              ws += N_NODES;
  float*    node_sum  = ws;              ws += N_NODES;
  unsigned* node_max  = (unsigned*)ws;   ws += N_NODES;
  float*    rel_logit = ws;              ws += 64;
  float*    elog      = ws;              ws += N_EDGES;

  init_kernel<<<2048, 256, 0, stream>>>(agg, node_sum, node_max, N_NODES * D, N_NODES);
  rel_kernel<<<1, 128, 0, stream>>>(W_r, a, rel_emb, rel_logit);
  gemm_kernel<<<(N_NODES + 63) / 64, 128, 0, stream>>>(X, W, Wres, bres, h, base);
  s_kernel<<<(N_NODES + 7) / 8, 256, 0, stream>>>(h, a, s_dst, s_src);
  logit_kernel<<<(N_EDGES + 255) / 256, 256, 0, stream>>>(src, dst, etype, s_dst, s_src,
                                                          rel_logit, elog, node_max);
  expsum_kernel<<<(N_EDGES + 255) / 256, 256, 0, stream>>>(dst, elog, node_max, node_sum);
  scatter_kernel<<<(N_EDGES + 7) / 8, 256, 0, stream>>>(src, dst, elog, node_sum, h, agg);
  final_kernel<<<(N_NODES * D + 255) / 256, 256, 0, stream>>>(agg, base, (float*)d_out);
}